// RepLKDepthwise_26293789786956
// MI455X (gfx1250) — compile-verified
//
#include <hip/hip_runtime.h>

// RepLK-style depthwise block: 13x13 dwconv + 3x3 dwconv + identity, each with
// folded inference BatchNorm, summed.  fp32, N=16 C=384 H=W=64.
//
// MI455X (gfx1250) strategy:
//  * one workgroup per (n,c) plane; the zero-padded plane lives in LDS
//    (76 rows x 80-float pitch = 24.3 KB of 320 KB) -> HBM traffic stays at
//    the ~201 MB read+write floor (~8.6 us at 23.3 TB/s).
//  * the op is VALU-compute-bound (~9 GFLOP fp32), so the 13x13 conv (94% of
//    FLOPs) runs on the matrix pipe: per kernel column dx it is a banded
//    Toeplitz matmul  out[y,x] = sum_t W[y,t]*X[t,x], W[m,t]=w[t-m,dx]
//    (13-wide band), 7 chained V_WMMA_F32_16X16X4_F32 (K=28) per 16x16 tile.
//  * the band is materialized as a ZERO-PADDED weight table wt[dx][dy+16] in
//    LDS, so the A operand is two unconditional adjacent ds loads -- no
//    compares / cndmask / exec-masked loads in the inner loop.
//  * each wave owns tiles (w) and (w+8): same x-range, y-ranges 32 rows apart,
//    so A is built ONCE and fed to two back-to-back WMMAs with two live
//    accumulators -> 182 WMMAs/wave, half the A traffic.
//  * interior input staging uses GLOBAL_LOAD_ASYNC_TO_LDS_B128 (direct
//    memory->LDS DMA, ASYNCcnt); only the halo border is zero-filled.
//  * 3x3 conv + identity + BN shifts run on the VALU epilogue; BN scales are
//    pre-folded into the LDS weight copies, shifts folded into one constant.

typedef float v2f __attribute__((ext_vector_type(2)));
typedef float v8f __attribute__((ext_vector_type(8)));

#define PW   80            // LDS row pitch (floats): keeps B128 async 16B-aligned
#define PH   76            // 64 + 2*6 rows
#define XOFF 8             // LDS column of global x==0 (left pad 8 for alignment)
#define YOFF 6             // LDS row    of global y==0
#define WTP  48            // padded Toeplitz table pitch: dy+16 in [1,43] always valid

__global__ __launch_bounds__(256)
void replk_dw_wmma(const float* __restrict__ x,
                   const float* __restrict__ w_large,
                   const float* __restrict__ w_small,
                   const float* __restrict__ g_l, const float* __restrict__ b_l,
                   const float* __restrict__ m_l, const float* __restrict__ v_l,
                   const float* __restrict__ g_s, const float* __restrict__ b_s,
                   const float* __restrict__ m_s, const float* __restrict__ v_s,
                   const float* __restrict__ g_i, const float* __restrict__ b_i,
                   const float* __restrict__ m_i, const float* __restrict__ v_i,
                   float* __restrict__ out, int C)
{
    __shared__ float lds_x[PH * PW];    // zero-padded input plane
    __shared__ float lds_wt[13 * WTP];  // padded Toeplitz table, BN-scaled
    __shared__ float lds_ws[9];         // 3x3 weights, BN-scaled
    __shared__ float lds_sc[6];         // {sc_l, sh_l, sc_s, sh_s, sc_i, sh_i}

    const int tid = threadIdx.x;
    const int nc  = blockIdx.x;            // plane index: n*C + c
    const int c   = nc % C;
    const float* __restrict__ xin = x + (size_t)nc * 64 * 64;

    // ---- phase 1: zero-fill halo border + weight table, launch async DMA --
    // 12 full border rows (y in [-6,-1] U [64,69]): 12*80 = 960 elements
    for (int i = tid; i < 960; i += 256) {
        int r  = i / 80;
        int xp = i - r * 80;
        int yp = (r < 6) ? r : (r + 64);
        lds_x[yp * PW + xp] = 0.0f;
    }
    // left/right pads of the 64 interior rows: 64*16 = 1024 elements
    for (int i = tid; i < 1024; i += 256) {
        int r  = i >> 4;
        int k  = i & 15;
        int xp = (k < 8) ? k : (k + 64);   // cols [0,7] and [72,79]
        lds_x[(r + YOFF) * PW + xp] = 0.0f;
    }
    // zero the padded Toeplitz table (scatter of live taps happens in phase 2)
    for (int i = tid; i < 13 * WTP; i += 256) lds_wt[i] = 0.0f;

    // interior: async DMA global -> LDS, 128b chunks (ASYNCcnt).
    // 64 rows x 16 chunks = 1024 B128 transfers, exactly 4 per thread; both
    // global and LDS addresses are 16B-aligned by construction.
    #pragma unroll
    for (int it = 0; it < 4; ++it) {
        const int q   = tid + it * 256;
        const int row = q >> 4;
        const int col = (q & 15) << 2;
        const float* gp = xin + row * 64 + col;
        unsigned loff = (unsigned)(size_t)(&lds_x[(row + YOFF) * PW + XOFF + col]);
        asm volatile("global_load_async_to_lds_b128 %0, %1, off"
                     :: "v"(loff), "v"(gp) : "memory");
    }

    // raw weights ride across the barrier in registers
    float rw = 0.0f, rs = 0.0f;
    if (tid < 169) {
        rw = w_large[c * 169 + tid];
    } else if (tid < 178) {
        rs = w_small[c * 9 + (tid - 169)];
    } else if (tid == 240) {
        float s = g_l[c] / sqrtf(v_l[c] + 1e-5f);
        lds_sc[0] = s; lds_sc[1] = b_l[c] - m_l[c] * s;
    } else if (tid == 241) {
        float s = g_s[c] / sqrtf(v_s[c] + 1e-5f);
        lds_sc[2] = s; lds_sc[3] = b_s[c] - m_s[c] * s;
    } else if (tid == 242) {
        float s = g_i[c] / sqrtf(v_i[c] + 1e-5f);
        lds_sc[4] = s; lds_sc[5] = b_i[c] - m_i[c] * s;
    }

    // async DMA completes on ASYNCcnt (not covered by the barrier's DS fence)
    asm volatile("s_wait_asynccnt 0x0" ::: "memory");
    __syncthreads();

    // ---- phase 2: scatter BN-scaled taps into the padded table -----------
    if (tid < 169) {
        const int dy = tid / 13;           // w_large is [dy][dx] row-major
        const int dx = tid - dy * 13;
        lds_wt[dx * WTP + dy + 16] = rw * lds_sc[0];
    } else if (tid < 178) {
        lds_ws[tid - 169] = rs * lds_sc[2];
    }
    __syncthreads();

    const int lane = tid & 31;
    const int wave = tid >> 5;
    const int m    = lane & 15;   // A: M row (output y-offset); B/D: N col (output x-offset)
    const int hi   = lane >> 4;   // half-wave select
    const int kh   = hi * 2;      // K sub-pair base (ISA 32-bit A/B layout)

    const float sc_i  = lds_sc[4];
    const float shift = lds_sc[1] + lds_sc[3] + lds_sc[5];

    // Wave w owns tiles w and w+8: ty in {0,1} and ty+2, same tx.
    // => identical Toeplitz A, B addresses differ by 32 LDS rows.
    const int y0 = (wave >> 2) * 16;       // upper tile; lower tile at y0+32
    const int x0 = (wave & 3)  * 16;

    v8f acc0 = {};                         // tile (y0,    x0)
    v8f acc1 = {};                         // tile (y0+32, x0)
    #pragma unroll
    for (int dx = 0; dx < 13; ++dx) {
        #pragma unroll
        for (int c7 = 0; c7 < 7; ++c7) {
            const int t0 = c7 * 4;
            const int ta = t0 + kh;        // global K index of this lane's .x element
            v2f a, b0, b1;
            // unconditional lookup in the zero-padded Toeplitz table:
            // index = dx*48 + (ta - m) + 16, always within [1, 43]
            const int wi = dx * WTP + (ta - m) + 16;
            a.x = lds_wt[wi];
            a.y = lds_wt[wi + 1];
            // X[t, n] = padded input at global row y0-6+t, col x0+n+dx-6
            const int bi = (y0 + ta) * PW + (x0 + m + dx + 2);   // -6 + XOFF = +2
            b0.x = lds_x[bi];
            b0.y = lds_x[bi + PW];
            b1.x = lds_x[bi + 32 * PW];
            b1.y = lds_x[bi + 33 * PW];
            acc0 = __builtin_amdgcn_wmma_f32_16x16x4_f32(
                       false, a, false, b0, (short)0, acc0, false, false);
            acc1 = __builtin_amdgcn_wmma_f32_16x16x4_f32(
                       false, a, false, b1, (short)0, acc1, false, false);
        }
    }

    // ---- epilogue: scaled 3x3 conv + identity + shifts, coalesced store --
    const int xg = x0 + m;                 // N dim -> x : lanes 0-15 contiguous
    #pragma unroll
    for (int j = 0; j < 2; ++j) {
        const int yb = y0 + 32 * j;
        #pragma unroll
        for (int r = 0; r < 8; ++r) {
            const int yg = yb + r + 8 * hi;    // M dim -> y
            float a3 = 0.0f;
            #pragma unroll
            for (int ky = 0; ky < 3; ++ky)
                #pragma unroll
                for (int kx = 0; kx < 3; ++kx)
                    a3 = fmaf(lds_ws[ky * 3 + kx],
                              lds_x[(yg + 5 + ky) * PW + (xg + 7 + kx)], a3);
            float acc = (j == 0) ? acc0[r] : acc1[r];
            float v = acc + a3 +
                      fmaf(sc_i, lds_x[(yg + YOFF) * PW + (xg + XOFF)], shift);
            out[((size_t)nc * 64 + yg) * 64 + xg] = v;
        }
    }
}

extern "C" void kernel_launch(void* const* d_in, const int* in_sizes, int n_in,
                              void* d_out, int out_size, void* d_ws, size_t ws_size,
                              hipStream_t stream) {
    (void)n_in; (void)out_size; (void)d_ws; (void)ws_size;
    const float* x       = (const float*)d_in[0];
    const float* w_large = (const float*)d_in[1];
    const float* w_small = (const float*)d_in[2];
    const float* g_l = (const float*)d_in[3];
    const float* b_l = (const float*)d_in[4];
    const float* m_l = (const float*)d_in[5];
    const float* v_l = (const float*)d_in[6];
    const float* g_s = (const float*)d_in[7];
    const float* b_s = (const float*)d_in[8];
    const float* m_s = (const float*)d_in[9];
    const float* v_s = (const float*)d_in[10];
    const float* g_i = (const float*)d_in[11];
    const float* b_i = (const float*)d_in[12];
    const float* m_i = (const float*)d_in[13];
    const float* v_i = (const float*)d_in[14];

    const int C      = in_sizes[3];              // 384 (gamma_l length)
    const int planes = in_sizes[0] / (64 * 64);  // N*C = 6144

    replk_dw_wmma<<<planes, 256, 0, stream>>>(
        x, w_large, w_small,
        g_l, b_l, m_l, v_l,
        g_s, b_s, m_s, v_s,
        g_i, b_i, m_i, v_i,
        (float*)d_out, C);
}